// EdgeGatedGraphConv_27290222198788
// MI455X (gfx1250) — compile-verified
//
#include <hip/hip_runtime.h>
#include <math.h>

// CDNA5 / gfx1250: wave32, WMMA f32 16x16x4.
typedef __attribute__((ext_vector_type(2))) float v2f;
typedef __attribute__((ext_vector_type(8))) float v8f;

#define DD 64
#define BN_EPS  1e-5f
#define GATE_EPS 1e-6f

__device__ __forceinline__ float fast_sigmoid(float x) {
    return 1.0f / (1.0f + __expf(-x));
}

__device__ __forceinline__ float atomic_add_f32(float* p, float v) {
    return __hip_atomic_fetch_add(p, v, __ATOMIC_RELAXED, __HIP_MEMORY_SCOPE_AGENT);
}

// ---------------------------------------------------------------------------
// Kernel 0: transpose the five 64x64 weight matrices into workspace so the
// WMMA B-fragment (two K-consecutive elements per lane) is a contiguous
// 8-byte load: WT[o*64 + i] = W[i*64 + o].
// Order in WT: [0]=W_sg [1]=W_dg [2]=W_du [3]=W_su [4]=W_eg
// ---------------------------------------------------------------------------
__global__ __launch_bounds__(256)
void transpose_w_kernel(const float* __restrict__ Wsg, const float* __restrict__ Wdg,
                        const float* __restrict__ Wdu, const float* __restrict__ Wsu,
                        const float* __restrict__ Weg, float* __restrict__ WT)
{
    const float* W;
    switch (blockIdx.x) {
      case 0:  W = Wsg; break;
      case 1:  W = Wdg; break;
      case 2:  W = Wdu; break;
      case 3:  W = Wsu; break;
      default: W = Weg; break;
    }
    float* dst = WT + (size_t)blockIdx.x * DD * DD;
    for (int idx = threadIdx.x; idx < DD * DD; idx += 256) {
        const int o = idx >> 6;      // out channel (row of WT)
        const int i = idx & 63;      // in channel  (col of WT)
        dst[idx] = W[i * DD + o];
    }
}

// ---------------------------------------------------------------------------
// Kernel 1: node GEMMs. Block = 128 threads (4 waves). Each wave handles one
// weight matrix for a 16-row node tile: out = nf[16x64] @ W[64x64] + b.
// WMMA f32 16x16x4, K-loop of 16 steps, 4 column tiles per wave.
// Layout (ISA 7.12.2):
//   A 16x4  : lane l holds row M=l&15; VGPR v <-> K = v + 2*(l>>4)
//   B 4x16  : lane l holds col N=l&15; VGPR v <-> K = v + 2*(l>>4)
//   C/D 16x16: lane l holds col N=l&15; VGPR v <-> M = v + 8*(l>>4)
// B fragment from transposed W: WT[col*64 + k + 2*kh .. +1] -> one b64 load.
// ---------------------------------------------------------------------------
__global__ __launch_bounds__(128)
void node_gemm_kernel(const float* __restrict__ nf,
                      const float* __restrict__ WT,
                      const float* __restrict__ B0, const float* __restrict__ B1,
                      const float* __restrict__ B2, const float* __restrict__ B3,
                      float* __restrict__ O0, float* __restrict__ O1,
                      float* __restrict__ O2, float* __restrict__ O3, int N)
{
    const int wave = threadIdx.x >> 5;
    const int lane = threadIdx.x & 31;
    const int nlo  = lane & 15;
    const int kh   = lane >> 4;
    const int row0 = blockIdx.x * 16;

    const float* bias; float* out;
    switch (wave) {
      case 0:  bias = B0; out = O0; break;
      case 1:  bias = B1; out = O1; break;
      case 2:  bias = B2; out = O2; break;
      default: bias = B3; out = O3; break;
    }
    const float* Wt = WT + (size_t)wave * DD * DD;

    v8f acc0 = {}, acc1 = {}, acc2 = {}, acc3 = {};
    const float* arow = nf + (size_t)(row0 + nlo) * DD;
    // Per-lane base of each column tile's row in WT (col-major weight rows).
    const float* w0 = Wt + (size_t)( 0 + nlo) * DD + 2 * kh;
    const float* w1 = Wt + (size_t)(16 + nlo) * DD + 2 * kh;
    const float* w2 = Wt + (size_t)(32 + nlo) * DD + 2 * kh;
    const float* w3 = Wt + (size_t)(48 + nlo) * DD + 2 * kh;

    #pragma unroll
    for (int k = 0; k < DD; k += 4) {
        v2f a  = *(const v2f*)(arow + k + 2 * kh);   // A: K = k+2kh, k+2kh+1
        v2f b0 = *(const v2f*)(w0 + k);
        v2f b1 = *(const v2f*)(w1 + k);
        v2f b2 = *(const v2f*)(w2 + k);
        v2f b3 = *(const v2f*)(w3 + k);
        acc0 = __builtin_amdgcn_wmma_f32_16x16x4_f32(false, a, false, b0, (short)0, acc0, false, false);
        acc1 = __builtin_amdgcn_wmma_f32_16x16x4_f32(false, a, false, b1, (short)0, acc1, false, false);
        acc2 = __builtin_amdgcn_wmma_f32_16x16x4_f32(false, a, false, b2, (short)0, acc2, false, false);
        acc3 = __builtin_amdgcn_wmma_f32_16x16x4_f32(false, a, false, b3, (short)0, acc3, false, false);
    }

#define STORE_TILE(ACC, T)                                                   \
    {                                                                        \
        const int col = (T) * 16 + nlo;                                      \
        const float bv = bias[col];                                          \
        _Pragma("unroll")                                                    \
        for (int v = 0; v < 8; ++v) {                                        \
            out[(size_t)(row0 + v + 8 * kh) * DD + col] = ACC[v] + bv;       \
        }                                                                    \
    }
    STORE_TILE(acc0, 0) STORE_TILE(acc1, 1) STORE_TILE(acc2, 2) STORE_TILE(acc3, 3)
#undef STORE_TILE
}

// ---------------------------------------------------------------------------
// Kernel 2: edge GEMM + gather + gated scatter.
// Block = 128 threads (4 waves), 16 edges per block, one 16x16 tile per wave.
// m = ef@W_eg + b_eg + Esrc[src] + Edst[dst];  sigma = sigmoid(m)
// Atomics: ssh[dst] += Bh[src]*sigma ; ss[dst] += sigma. m stored to d_out.
// ---------------------------------------------------------------------------
__global__ __launch_bounds__(128)
void edge_gemm_kernel(const float* __restrict__ ef,
                      const long long* __restrict__ srcIdx,
                      const long long* __restrict__ dstIdx,
                      const float* __restrict__ WTeg, const float* __restrict__ beg,
                      const float* __restrict__ Esrc, const float* __restrict__ Edst,
                      const float* __restrict__ BhM,
                      float* __restrict__ mOut,
                      float* __restrict__ ssh, float* __restrict__ ss, int E)
{
    __shared__ int s_src[16];
    __shared__ int s_dst[16];
    const int wave = threadIdx.x >> 5;
    const int lane = threadIdx.x & 31;
    const int nlo  = lane & 15;
    const int kh   = lane >> 4;
    const int row0 = blockIdx.x * 16;

    if (threadIdx.x < 16)       s_src[threadIdx.x]      = (int)srcIdx[row0 + threadIdx.x];
    else if (threadIdx.x < 32)  s_dst[threadIdx.x - 16] = (int)dstIdx[row0 + threadIdx.x - 16];
    __syncthreads();

    // Warm the gathered rows into cache (lowers to global_prefetch_b8).
    if (threadIdx.x < 16) {
        __builtin_prefetch(Esrc + (size_t)s_src[threadIdx.x] * DD, 0, 3);
        __builtin_prefetch(BhM  + (size_t)s_src[threadIdx.x] * DD, 0, 3);
    } else if (threadIdx.x < 32) {
        __builtin_prefetch(Edst + (size_t)s_dst[threadIdx.x - 16] * DD, 0, 3);
    }

    const int n0  = wave * 16;
    const int col = n0 + nlo;
    v8f acc = {};
    const float* arow = ef + (size_t)(row0 + nlo) * DD;
    const float* wrow = WTeg + (size_t)col * DD + 2 * kh;

    #pragma unroll
    for (int k = 0; k < DD; k += 4) {
        v2f a = *(const v2f*)(arow + k + 2 * kh);
        v2f b = *(const v2f*)(wrow + k);
        acc = __builtin_amdgcn_wmma_f32_16x16x4_f32(false, a, false, b, (short)0, acc, false, false);
    }

    const float bc = beg[col];
    #pragma unroll
    for (int v = 0; v < 8; ++v) {
        const int row = v + 8 * kh;
        const int s = s_src[row];
        const int d = s_dst[row];
        float mval = acc[v] + bc + Esrc[(size_t)s * DD + col] + Edst[(size_t)d * DD + col];
        mOut[(size_t)(row0 + row) * DD + col] = mval;
        float sig = fast_sigmoid(mval);
        float bh  = BhM[(size_t)s * DD + col];
        atomic_add_f32(&ssh[(size_t)d * DD + col], bh * sig);
        atomic_add_f32(&ss [(size_t)d * DD + col], sig);
    }
}

// ---------------------------------------------------------------------------
// Kernel 3: x = Xlin + ssh/(ss+eps) (in place in d_out) + BN stat reduction.
// Grid-stride; column = tid&63 is invariant. LDS ds_add_f32 block reduction,
// then 128 global atomics per block into stats[0:64]=sum, stats[64:128]=sumsq.
// ---------------------------------------------------------------------------
__global__ __launch_bounds__(256)
void node_pre_stats_kernel(float* __restrict__ xout,
                           const float* __restrict__ ssh, const float* __restrict__ ss,
                           float* __restrict__ stats, int total)
{
    __shared__ float lsum[DD];
    __shared__ float lsq[DD];
    if (threadIdx.x < DD) { lsum[threadIdx.x] = 0.f; lsq[threadIdx.x] = 0.f; }
    __syncthreads();

    const int c = threadIdx.x & (DD - 1);
    float psum = 0.f, psq = 0.f;
    for (size_t i = (size_t)blockIdx.x * blockDim.x + threadIdx.x; i < (size_t)total;
         i += (size_t)gridDim.x * blockDim.x) {
        float x = xout[i] + ssh[i] / (ss[i] + GATE_EPS);
        xout[i] = x;
        psum += x; psq += x * x;
    }
    atomicAdd(&lsum[c], psum);   // ds_add_f32
    atomicAdd(&lsq[c],  psq);
    __syncthreads();
    if (threadIdx.x < DD) {
        atomic_add_f32(&stats[threadIdx.x],      lsum[threadIdx.x]);
        atomic_add_f32(&stats[DD + threadIdx.x], lsq[threadIdx.x]);
    }
}

// Same reduction, no rewrite (edge stats over m).
__global__ __launch_bounds__(256)
void edge_stats_kernel(const float* __restrict__ m, float* __restrict__ stats, int total)
{
    __shared__ float lsum[DD];
    __shared__ float lsq[DD];
    if (threadIdx.x < DD) { lsum[threadIdx.x] = 0.f; lsq[threadIdx.x] = 0.f; }
    __syncthreads();

    const int c = threadIdx.x & (DD - 1);
    float psum = 0.f, psq = 0.f;
    for (size_t i = (size_t)blockIdx.x * blockDim.x + threadIdx.x; i < (size_t)total;
         i += (size_t)gridDim.x * blockDim.x) {
        float x = m[i];
        psum += x; psq += x * x;
    }
    atomicAdd(&lsum[c], psum);
    atomicAdd(&lsq[c],  psq);
    __syncthreads();
    if (threadIdx.x < DD) {
        atomic_add_f32(&stats[threadIdx.x],      lsum[threadIdx.x]);
        atomic_add_f32(&stats[DD + threadIdx.x], lsq[threadIdx.x]);
    }
}

// ---------------------------------------------------------------------------
// Kernel 4: out = resid + silu(batchnorm(out)) in place (nodes or edges).
// ---------------------------------------------------------------------------
__global__ __launch_bounds__(256)
void finalize_kernel(float* __restrict__ out, const float* __restrict__ resid,
                     const float* __restrict__ stats,
                     const float* __restrict__ gamma, const float* __restrict__ beta,
                     float invCount, int total)
{
    for (size_t i = (size_t)blockIdx.x * blockDim.x + threadIdx.x; i < (size_t)total;
         i += (size_t)gridDim.x * blockDim.x) {
        const int c = (int)(i & (DD - 1));
        float mean = stats[c] * invCount;
        float var  = stats[DD + c] * invCount - mean * mean;
        float xn   = (out[i] - mean) * rsqrtf(var + BN_EPS) * gamma[c] + beta[c];
        out[i] = resid[i] + xn * fast_sigmoid(xn);
    }
}

// ---------------------------------------------------------------------------
extern "C" void kernel_launch(void* const* d_in, const int* in_sizes, int n_in,
                              void* d_out, int out_size, void* d_ws, size_t ws_size,
                              hipStream_t stream)
{
    const long long* edge_index = (const long long*)d_in[0];
    const float* nf  = (const float*)d_in[1];
    const float* ef  = (const float*)d_in[2];
    const float* Wsg = (const float*)d_in[3];  const float* bsg = (const float*)d_in[4];
    const float* Wdg = (const float*)d_in[5];  const float* bdg = (const float*)d_in[6];
    const float* Weg = (const float*)d_in[7];  const float* beg = (const float*)d_in[8];
    const float* Wsu = (const float*)d_in[9];  const float* bsu = (const float*)d_in[10];
    const float* Wdu = (const float*)d_in[11]; const float* bdu = (const float*)d_in[12];
    const float* gn  = (const float*)d_in[13]; const float* btn = (const float*)d_in[14];
    const float* ge  = (const float*)d_in[15]; const float* bte = (const float*)d_in[16];

    const int E = in_sizes[0] / 2;
    const int N = in_sizes[1] / DD;
    const long long* src = edge_index;       // edge_index[0]
    const long long* dst = edge_index + E;   // edge_index[1]

    const size_t ND = (size_t)N * DD;
    const size_t ED = (size_t)E * DD;

    // Workspace layout (floats): Esrc | Edst | Bh | ssh | ss | stats(256) | WT(5*4096)
    float* ws    = (float*)d_ws;
    float* Esrc  = ws;
    float* Edst  = ws + ND;
    float* Bh    = ws + 2 * ND;
    float* ssh   = ws + 3 * ND;
    float* ss    = ws + 4 * ND;
    float* stats = ws + 5 * ND;          // [0:64) nsum [64:128) nsq [128:192) esum [192:256) esq
    float* WT    = ws + 5 * ND + 256;    // 5 transposed 64x64 weight matrices

    float* xout = (float*)d_out;   // x region, used as Xlin scratch then final x
    float* mOut = xout + ND;       // y region, holds m then final y

    hipMemsetAsync(ssh, 0, (2 * ND + 256) * sizeof(float), stream);

    transpose_w_kernel<<<5, 256, 0, stream>>>(Wsg, Wdg, Wdu, Wsu, Weg, WT);

    node_gemm_kernel<<<(N + 15) / 16, 128, 0, stream>>>(
        nf, WT, bsg, bdg, bdu, bsu,
        Esrc, Edst, Bh, xout, N);

    edge_gemm_kernel<<<(E + 15) / 16, 128, 0, stream>>>(
        ef, src, dst, WT + 4 * DD * DD, beg, Esrc, Edst, Bh, mOut, ssh, ss, E);

    node_pre_stats_kernel<<<1024, 256, 0, stream>>>(xout, ssh, ss, stats, (int)ND);
    edge_stats_kernel<<<2048, 256, 0, stream>>>(mOut, stats + 128, (int)ED);

    finalize_kernel<<<1024, 256, 0, stream>>>(xout, nf, stats,       gn, btn, 1.0f / (float)N, (int)ND);
    finalize_kernel<<<2048, 256, 0, stream>>>(mOut, ef, stats + 128, ge, bte, 1.0f / (float)E, (int)ED);
}